// IntensityCellReject_67087389164244
// MI455X (gfx1250) — compile-verified
//
#include <hip/hip_runtime.h>
#include <stdint.h>

// IntensityCellReject: segment mean per label, threshold, masked rewrite.
// N = 512*512*64 = 16,777,216 ; labels in [0, 512)

#define NUM_LABELS 512
#define NPART      16          // partial accumulator arrays in d_ws
#define ACC_BLOCKS 1024
#define ACC_THREADS 256
#define DEPTH      4           // async pipeline depth (vec4 tiles in flight)

typedef float vfloat4 __attribute__((ext_vector_type(4)));  // native vec for NT store

// ---------------- kernel 0: zero workspace partials -------------------------
__global__ void zero_ws_kernel(float* __restrict__ ws, int n) {
    int i = blockIdx.x * blockDim.x + threadIdx.x;
    if (i < n) ws[i] = 0.0f;
}

// ---------------- kernel 1: per-block LDS histogram, async b128 streamed ----
__global__ __launch_bounds__(ACC_THREADS)
void accumulate_kernel(const int* __restrict__ mask,
                       const float* __restrict__ image,
                       float* __restrict__ partial,
                       unsigned N) {
    __shared__ float  s_sum[NUM_LABELS];
    __shared__ float  s_cnt[NUM_LABELS];
    __shared__ int4   s_m4[DEPTH][ACC_THREADS];   // 16 KB
    __shared__ float4 s_v4[DEPTH][ACC_THREADS];   // 16 KB

    const int tid = threadIdx.x;
    for (int l = tid; l < NUM_LABELS; l += ACC_THREADS) {
        s_sum[l] = 0.0f;
        s_cnt[l] = 0.0f;
    }
    __syncthreads();

    const unsigned stride  = gridDim.x * blockDim.x;        // in vec4 units
    const unsigned idx0    = blockIdx.x * blockDim.x + tid; // in vec4 units
    const unsigned vchunk  = stride * 4u;                   // elements per sweep

    if ((N % vchunk) == 0) {
        // Uniform trip count, 16B-aligned vec4 tiles => async b128 pipeline.
        const unsigned K = N / vchunk;                      // tiles/thread (16)
        const unsigned P = (K < DEPTH) ? K : DEPTH;

        const int4*   mbase = (const int4*)mask;
        const float4* vbase = (const float4*)image;

        // Prologue: fill the pipeline.
        for (unsigned p = 0; p < P; ++p) {
            const unsigned g = idx0 + p * stride;
            uint32_t lm = (uint32_t)(uintptr_t)&s_m4[p & (DEPTH - 1)][tid];
            uint32_t lv = (uint32_t)(uintptr_t)&s_v4[p & (DEPTH - 1)][tid];
            const int4*   ma = mbase + g;
            const float4* va = vbase + g;
            asm volatile("global_load_async_to_lds_b128 %0, %1, off"
                         :: "v"(lm), "v"(ma) : "memory");
            asm volatile("global_load_async_to_lds_b128 %0, %1, off th:TH_LOAD_NT"
                         :: "v"(lv), "v"(va) : "memory");
        }

        unsigned k = 0;
        // Steady state: outstanding = DEPTH tiles = 2*DEPTH ops; waiting to
        // <= 2*DEPTH-2 guarantees the oldest tile landed (in-order returns).
        for (; k + DEPTH < K; ++k) {
            asm volatile("s_wait_asynccnt 6" ::: "memory");
            {
                int4   m = s_m4[k & (DEPTH - 1)][tid];
                float4 v = s_v4[k & (DEPTH - 1)][tid];
                atomicAdd(&s_sum[m.x], v.x); atomicAdd(&s_cnt[m.x], 1.0f);
                atomicAdd(&s_sum[m.y], v.y); atomicAdd(&s_cnt[m.y], 1.0f);
                atomicAdd(&s_sum[m.z], v.z); atomicAdd(&s_cnt[m.z], 1.0f);
                atomicAdd(&s_sum[m.w], v.w); atomicAdd(&s_cnt[m.w], 1.0f);
            }
            const unsigned p = k + DEPTH;
            const unsigned g = idx0 + p * stride;
            uint32_t lm = (uint32_t)(uintptr_t)&s_m4[p & (DEPTH - 1)][tid];
            uint32_t lv = (uint32_t)(uintptr_t)&s_v4[p & (DEPTH - 1)][tid];
            const int4*   ma = mbase + g;
            const float4* va = vbase + g;
            asm volatile("global_load_async_to_lds_b128 %0, %1, off"
                         :: "v"(lm), "v"(ma) : "memory");
            asm volatile("global_load_async_to_lds_b128 %0, %1, off th:TH_LOAD_NT"
                         :: "v"(lv), "v"(va) : "memory");
        }
        // Drain.
        asm volatile("s_wait_asynccnt 0" ::: "memory");
        for (; k < K; ++k) {
            int4   m = s_m4[k & (DEPTH - 1)][tid];
            float4 v = s_v4[k & (DEPTH - 1)][tid];
            atomicAdd(&s_sum[m.x], v.x); atomicAdd(&s_cnt[m.x], 1.0f);
            atomicAdd(&s_sum[m.y], v.y); atomicAdd(&s_cnt[m.y], 1.0f);
            atomicAdd(&s_sum[m.z], v.z); atomicAdd(&s_cnt[m.z], 1.0f);
            atomicAdd(&s_sum[m.w], v.w); atomicAdd(&s_cnt[m.w], 1.0f);
        }
    } else {
        // Generic fallback: plain global loads, any N.
        for (unsigned i = idx0; i < N; i += stride) {
            int   m = mask[i];
            float v = image[i];
            atomicAdd(&s_sum[m], v);
            atomicAdd(&s_cnt[m], 1.0f);
        }
    }

    __syncthreads();
    // Flush block histogram into one of NPART partial arrays (low contention).
    float* dst = partial + (unsigned)(blockIdx.x & (NPART - 1)) * (2 * NUM_LABELS);
    for (int l = tid; l < NUM_LABELS; l += ACC_THREADS) {
        atomicAdd(&dst[l],              s_sum[l]);
        atomicAdd(&dst[NUM_LABELS + l], s_cnt[l]);
    }
}

// ---------------- kernel 2: fold partials -> keep[512] ----------------------
__global__ void compute_keep_kernel(const float* __restrict__ partial,
                                    float* __restrict__ keep) {
    const int l = threadIdx.x;                  // 512 threads, 1 block
    float s = 0.0f, c = 0.0f;
    #pragma unroll
    for (int p = 0; p < NPART; ++p) {
        s += partial[p * (2 * NUM_LABELS) + l];
        c += partial[p * (2 * NUM_LABELS) + NUM_LABELS + l];
    }
    const float cutoff = (0.07f - 0.5f) / 0.5f; // -0.86
    float mean = s / fmaxf(c, 1.0f);
    keep[l] = (mean >= cutoff) ? 1.0f : 0.0f;
}

// ---------------- kernel 3: out[i] = mask[i] * keep[mask[i]] ----------------
__global__ __launch_bounds__(256)
void apply_kernel(const int* __restrict__ mask,
                  const float* __restrict__ keep,
                  float* __restrict__ out,
                  unsigned N) {
    __shared__ float s_keep[NUM_LABELS];
    for (int l = threadIdx.x; l < NUM_LABELS; l += blockDim.x)
        s_keep[l] = keep[l];
    __syncthreads();

    const unsigned stride = gridDim.x * blockDim.x;
    const unsigned t0     = blockIdx.x * blockDim.x + threadIdx.x;

    if ((N & 3u) == 0u) {
        const int4*    m4 = (const int4*)mask;
        vfloat4*       o4 = (vfloat4*)out;
        const unsigned NV = N >> 2;
        for (unsigned i = t0; i < NV; i += stride) {
            int4 m = m4[i];
            vfloat4 o;
            o.x = (float)m.x * s_keep[m.x];
            o.y = (float)m.y * s_keep[m.y];
            o.z = (float)m.z * s_keep[m.z];
            o.w = (float)m.w * s_keep[m.w];
            __builtin_nontemporal_store(o, &o4[i]);
        }
    } else {
        for (unsigned i = t0; i < N; i += stride) {
            int m = mask[i];
            out[i] = (float)m * s_keep[m];
        }
    }
}

// ---------------- launch ----------------------------------------------------
extern "C" void kernel_launch(void* const* d_in, const int* in_sizes, int n_in,
                              void* d_out, int out_size, void* d_ws, size_t ws_size,
                              hipStream_t stream) {
    const int*   mask  = (const int*)d_in[0];
    const float* image = (const float*)d_in[1];
    float*       out   = (float*)d_out;
    unsigned     N     = (unsigned)in_sizes[0];

    float* ws      = (float*)d_ws;
    float* partial = ws;                              // NPART * 2 * 512 floats
    float* keep    = ws + NPART * 2 * NUM_LABELS;     // 512 floats

    const int zn = NPART * 2 * NUM_LABELS;
    zero_ws_kernel<<<(zn + 255) / 256, 256, 0, stream>>>(ws, zn);

    accumulate_kernel<<<ACC_BLOCKS, ACC_THREADS, 0, stream>>>(mask, image, partial, N);

    compute_keep_kernel<<<1, NUM_LABELS, 0, stream>>>(partial, keep);

    apply_kernel<<<2048, 256, 0, stream>>>(mask, keep, out, N);
}